// GATConv_88656714924157
// MI455X (gfx1250) — compile-verified
//
#include <hip/hip_runtime.h>

#define N_NODES   100000
#define IN_CH     256
#define OUT_CH    128
#define NEG_SLOPE 0.2f
#define EPS_F     1e-16f

typedef __attribute__((ext_vector_type(2))) float v2f;
typedef __attribute__((ext_vector_type(8))) float v8f;

// Monotonic float<->uint encoding so atomicMax(unsigned) orders floats.
__device__ __forceinline__ unsigned enc_f32(float f) {
  unsigned u = __float_as_uint(f);
  return (u & 0x80000000u) ? ~u : (u | 0x80000000u);
}
__device__ __forceinline__ float dec_f32(unsigned e) {
  return __uint_as_float((e & 0x80000000u) ? (e ^ 0x80000000u) : ~e);
}
__device__ __forceinline__ float leaky(float a) {
  return a > 0.0f ? a : NEG_SLOPE * a;
}

// ---------------------------------------------------------------------------
// Kernel 1: xp = x @ W  using V_WMMA_F32_16X16X4_F32.
// One wave -> 16 output rows x 128 cols (8 accumulator tiles of 16x16).
// A layout (16x4 f32): lanes 0-15 M=lane, K={k,k+1}; lanes 16-31 same M, K={k+2,k+3}.
// B layout (4x16 f32): VGPR0 = K0|K2, VGPR1 = K1|K3, N = lane%16.
// ---------------------------------------------------------------------------
__global__ __launch_bounds__(256) void gat_gemm(const float* __restrict__ x,
                                                const float* __restrict__ w,
                                                float* __restrict__ xp) {
  const int wave = (int)((blockIdx.x * blockDim.x + threadIdx.x) >> 5);
  const int lane = (int)(threadIdx.x & 31);
  if (wave >= N_NODES / 16) return;          // uniform per wave: EXEC stays all-1
  const int laneLo  = lane & 15;
  const int kh      = (lane >> 4) << 1;      // 0 for lanes 0-15, 2 for lanes 16-31
  const int rowBase = wave << 4;

  const float* arow = x + (size_t)(rowBase + laneLo) * IN_CH + kh;
  v8f acc[8] = {};

#pragma unroll 4
  for (int k = 0; k < IN_CH; k += 4) {
    v2f a = *(const v2f*)(arow + k);                       // K={k+kh, k+kh+1}
    const float* bp = w + (size_t)(k + kh) * OUT_CH + laneLo;
#pragma unroll
    for (int t = 0; t < 8; ++t) {
      v2f b;
      b.x = bp[t * 16];                                    // W[k+kh  ][col]
      b.y = bp[t * 16 + OUT_CH];                           // W[k+kh+1][col]
      acc[t] = __builtin_amdgcn_wmma_f32_16x16x4_f32(
          /*neg_a=*/false, a, /*neg_b=*/false, b,
          /*c_mod=*/(short)0, acc[t], /*reuse_a=*/false, /*reuse_b=*/false);
    }
  }

  // C/D layout: VGPR r -> M = rowBase + r + 8*(lane>>4), N = lane%16
  const int row0 = rowBase + ((lane >> 4) << 3);
#pragma unroll
  for (int t = 0; t < 8; ++t) {
    const int col = t * 16 + laneLo;
#pragma unroll
    for (int r = 0; r < 8; ++r)
      xp[(size_t)(row0 + r) * OUT_CH + col] = acc[t][r];
  }
}

// ---------------------------------------------------------------------------
// Kernel 2: per-node logits alpha_src/alpha_dst = xp . att; also zero denom
// and initialize the global-max cell.
// ---------------------------------------------------------------------------
__global__ __launch_bounds__(256) void node_alpha(const float* __restrict__ xp,
                                                  const float* __restrict__ att_s,
                                                  const float* __restrict__ att_d,
                                                  float* __restrict__ as,
                                                  float* __restrict__ ad,
                                                  float* __restrict__ denom,
                                                  unsigned* __restrict__ amax) {
  const int i = blockIdx.x * blockDim.x + threadIdx.x;
  if (i == 0) *amax = 0x007FFFFFu;           // enc(-inf)
  if (i >= N_NODES) return;
  const float4* row = (const float4*)(xp + (size_t)i * OUT_CH);
  const float4* a4  = (const float4*)att_s;
  const float4* b4  = (const float4*)att_d;
  float s = 0.0f, d = 0.0f;
#pragma unroll 4
  for (int c = 0; c < OUT_CH / 4; ++c) {
    float4 v = row[c], a = a4[c], b = b4[c];
    s += v.x * a.x + v.y * a.y + v.z * a.z + v.w * a.w;
    d += v.x * b.x + v.y * b.y + v.z * b.z + v.w * b.w;
  }
  as[i] = s; ad[i] = d; denom[i] = 0.0f;
}

// ---------------------------------------------------------------------------
// Kernel 3: global max over leaky_relu(as[src]+ad[dst]) for valid edges and
// self loops. Wave-level shuffle reduction, then one atomicMax per wave.
// ---------------------------------------------------------------------------
__global__ __launch_bounds__(256) void edge_max(const long long* __restrict__ src,
                                                const long long* __restrict__ dst,
                                                long long E,
                                                const float* __restrict__ as,
                                                const float* __restrict__ ad,
                                                unsigned* __restrict__ amax) {
  const long long e = (long long)blockIdx.x * blockDim.x + threadIdx.x;
  float v = -__builtin_inff();
  if (e < E) {
    const long long s = src[e], d = dst[e];
    if (s != d) v = leaky(as[s] + ad[d]);
  } else if (e < E + N_NODES) {
    const long long i = e - E;
    v = leaky(as[i] + ad[i]);
  }
#pragma unroll
  for (int off = 16; off > 0; off >>= 1)
    v = fmaxf(v, __shfl_xor(v, off, 32));
  if ((threadIdx.x & 31) == 0 && v > -__builtin_inff())
    atomicMax(amax, enc_f32(v));
}

// ---------------------------------------------------------------------------
// Kernel 4: ex_e = exp(alpha_e - amax); accumulate denom[dst] atomically.
// Self loops (e >= E) write exs[i] and add to denom[i].
// ---------------------------------------------------------------------------
__global__ __launch_bounds__(256) void edge_exp(const long long* __restrict__ src,
                                                const long long* __restrict__ dst,
                                                long long E,
                                                const float* __restrict__ as,
                                                const float* __restrict__ ad,
                                                const unsigned* __restrict__ amax_u,
                                                float* __restrict__ ex,
                                                float* __restrict__ exs,
                                                float* __restrict__ denom) {
  const long long e = (long long)blockIdx.x * blockDim.x + threadIdx.x;
  if (e >= E + N_NODES) return;
  const float amax = dec_f32(*amax_u);
  if (e < E) {
    const long long s = src[e], d = dst[e];
    if (s == d) { ex[e] = 0.0f; return; }
    const float wgt = expf(leaky(as[s] + ad[d]) - amax);
    ex[e] = wgt;
    atomicAdd(&denom[d], wgt);
  } else {
    const long long i = e - E;
    const float wgt = expf(leaky(as[i] + ad[i]) - amax);
    exs[i] = wgt;
    atomicAdd(&denom[i], wgt);
  }
}

// ---------------------------------------------------------------------------
// Kernel 5: out[i] = xp[i] * (exs[i]/(denom[i]+eps)) + bias  (self-loop term,
// fully initializes d_out before the atomic scatter).
// ---------------------------------------------------------------------------
__global__ __launch_bounds__(OUT_CH) void self_bias(const float* __restrict__ xp,
                                                    const float* __restrict__ exs,
                                                    const float* __restrict__ denom,
                                                    const float* __restrict__ bias,
                                                    float* __restrict__ out) {
  const int i = blockIdx.x;
  const int c = threadIdx.x;
  const float coef = exs[i] / (denom[i] + EPS_F);
  out[(size_t)i * OUT_CH + c] = xp[(size_t)i * OUT_CH + c] * coef + bias[c];
}

// ---------------------------------------------------------------------------
// Kernel 6: scatter-aggregate. One wave per edge; lane L handles columns
// 4L..4L+3 via a float4 gather of xp[src] and 4 global fp32 atomic adds.
// xp and out both fit in the 192MB L2, so gathers + atomics stay L2-resident.
// ---------------------------------------------------------------------------
__global__ __launch_bounds__(256) void scatter_agg(const long long* __restrict__ src,
                                                   const long long* __restrict__ dst,
                                                   long long E,
                                                   const float* __restrict__ xp,
                                                   const float* __restrict__ ex,
                                                   const float* __restrict__ denom,
                                                   float* __restrict__ out) {
  const long long wave = ((long long)blockIdx.x * blockDim.x + threadIdx.x) >> 5;
  const int lane = (int)(threadIdx.x & 31);
  if (wave >= E) return;
  const float wgt = ex[wave];
  if (wgt == 0.0f) return;                   // masked (self) edge
  const long long s = src[wave], d = dst[wave];
  const float coef = wgt / (denom[d] + EPS_F);
  const float4 v = ((const float4*)(xp + (size_t)s * OUT_CH))[lane];
  float* o = out + (size_t)d * OUT_CH + lane * 4;
  atomicAdd(o + 0, v.x * coef);
  atomicAdd(o + 1, v.y * coef);
  atomicAdd(o + 2, v.z * coef);
  atomicAdd(o + 3, v.w * coef);
}

// ---------------------------------------------------------------------------
extern "C" void kernel_launch(void* const* d_in, const int* in_sizes, int n_in,
                              void* d_out, int out_size, void* d_ws, size_t ws_size,
                              hipStream_t stream) {
  const float*     x     = (const float*)d_in[0];
  const long long* ei    = (const long long*)d_in[1];   // int64 [2, E]
  const float*     w     = (const float*)d_in[2];
  const float*     att_s = (const float*)d_in[3];
  const float*     att_d = (const float*)d_in[4];
  const float*     bias  = (const float*)d_in[5];
  const long long  E     = (long long)in_sizes[1] / 2;
  const long long* src   = ei;
  const long long* dst   = ei + E;
  float* out = (float*)d_out;

  // Workspace layout (all re-initialized every call; ~59 MB total).
  char* p = (char*)d_ws;
  float* xp    = (float*)p;  p += (size_t)N_NODES * OUT_CH * sizeof(float);
  float* as    = (float*)p;  p += (size_t)N_NODES * sizeof(float);
  float* ad    = (float*)p;  p += (size_t)N_NODES * sizeof(float);
  float* denom = (float*)p;  p += (size_t)N_NODES * sizeof(float);
  float* exs   = (float*)p;  p += (size_t)N_NODES * sizeof(float);
  float* ex    = (float*)p;  p += (size_t)E * sizeof(float);
  unsigned* amax = (unsigned*)p;

  const int rowTiles = N_NODES / 16;                    // 6250, exact
  gat_gemm  <<<(rowTiles + 7) / 8, 256, 0, stream>>>(x, w, xp);
  node_alpha<<<(N_NODES + 255) / 256, 256, 0, stream>>>(xp, att_s, att_d, as, ad, denom, amax);
  const int edgeBlocks = (int)((E + N_NODES + 255) / 256);
  edge_max  <<<edgeBlocks, 256, 0, stream>>>(src, dst, E, as, ad, amax);
  edge_exp  <<<edgeBlocks, 256, 0, stream>>>(src, dst, E, as, ad, amax, ex, exs, denom);
  self_bias <<<N_NODES, OUT_CH, 0, stream>>>(xp, exs, denom, bias, out);
  scatter_agg<<<(int)((E * 32 + 255) / 256), 256, 0, stream>>>(src, dst, E, xp, ex, denom, out);
}